// CapsuleLayer_43877385895952
// MI455X (gfx1250) — compile-verified
//
#include <hip/hip_runtime.h>

typedef __attribute__((ext_vector_type(2))) float v2f;
typedef __attribute__((ext_vector_type(8))) float v8f;

#define BATCH 64
#define NI    2304
#define DIN   8
#define NJ    32
#define NE    16
#define JE    (NJ * NE)   // 512
#define BT    16          // batch tile (WMMA M)
#define IT    16          // capsules i per block
#define USTR  17          // padded e-stride to avoid LDS bank conflicts

// d_ws layout: S [BATCH*JE] f32, then Vacc [BATCH*JE] f32  (256 KB total)

__global__ void caps_zero(float* __restrict__ p, int n) {
    int i = blockIdx.x * blockDim.x + threadIdx.x;
    if (i < n) p[i] = 0.0f;
}

// One block: 16 batches (blockIdx.y) x 16 capsules-i (blockIdx.x), 256 threads = 8 waves.
// Per i: Phase A computes u_hat tile (16x512) with WMMA f32 16x16x4, stages to LDS.
//        Phase B computes logits = b0 + u_hat.Vacc, softmax over j, accumulates S in regs.
// End: atomic-add partial S into global S.
__global__ __launch_bounds__(256) void caps_partial(
    const float* __restrict__ x,    // [B, I, DIN]
    const float* __restrict__ W,    // [I, J, DIN, E]
    const float* __restrict__ b0,   // [I, J]
    const float* __restrict__ Vacc, // [B, J, E]  (sum of V from previous iterations)
    float* __restrict__ Sg)         // [B, J, E]  (accumulated with atomics)
{
    __shared__ float U [BT * NJ * USTR];  // u_hat tile, padded
    __shared__ float Vl[BT * NJ * USTR];  // staged Vacc,  padded

    const int tid   = threadIdx.x;
    const int bbase = blockIdx.y * BT;
    const int i0    = blockIdx.x * IT;

    // ---- stage Vacc for this batch tile into padded LDS ----
    for (int k = tid; k < BT * JE; k += 256) {
        int b = k >> 9;            // /512
        int r = k & 511;
        int j = r >> 4;
        int e = r & 15;
        Vl[(b * NJ + j) * USTR + e] = Vacc[(size_t)(bbase + b) * JE + r];
    }
    __syncthreads();

    // Phase-A lane mapping (WMMA f32 16x16x4 layouts)
    const int lane = tid & 31;
    const int wave = tid >> 5;     // 0..7, each wave owns j = wave + 8*p
    const int m16  = lane & 15;    // M (batch row) / N (e column)
    const int hi   = lane >> 4;    // K-half selector

    // Phase-B thread mapping: thread -> (batch bb, capsule pair j0,j1)
    const int bb = tid >> 4;          // 0..15
    const int j0 = (tid & 15) * 2;
    const int j1 = j0 + 1;

    float s0[NE], s1[NE];
    #pragma unroll
    for (int e = 0; e < NE; ++e) { s0[e] = 0.0f; s1[e] = 0.0f; }

    for (int ii = 0; ii < IT; ++ii) {
        const int i = i0 + ii;

        // ---------------- Phase A: u_hat = X(16x8) @ W_i(8x512) ----------------
        // A layout (16x4 f32): lane<16 -> M=lane, K={0,1}; lane>=16 -> M=lane-16, K={2,3}
        const float* xp = x + ((size_t)(bbase + m16) * NI + i) * DIN + hi * 2;
        v2f a_lo = *(const v2f*)(xp);       // K = 0..3 half
        v2f a_hi = *(const v2f*)(xp + 4);   // K = 4..7 half

        #pragma unroll
        for (int p = 0; p < 4; ++p) {
            const int j = wave + p * 8;
            const float* wp = W + (((size_t)i * NJ + j) * DIN) * NE + m16;
            // prefetch next capsule's W slice (speculative; dropped if invalid)
            __builtin_prefetch(wp + NJ * DIN * NE, 0, 1);
            // B layout (4x16 f32): lane<16 -> K={0,1}, N=lane; lane>=16 -> K={2,3}
            v2f bl = { wp[hi * 32],      wp[hi * 32 + 16] };      // K 0..3
            v2f bh = { wp[64 + hi * 32], wp[64 + hi * 32 + 16] }; // K 4..7
            v8f acc = {0.f, 0.f, 0.f, 0.f, 0.f, 0.f, 0.f, 0.f};
            acc = __builtin_amdgcn_wmma_f32_16x16x4_f32(
                      false, a_lo, false, bl, (short)0, acc, false, false);
            acc = __builtin_amdgcn_wmma_f32_16x16x4_f32(
                      false, a_hi, false, bh, (short)0, acc, false, false);
            // D layout: lane<16 -> N=lane, M=r; lane>=16 -> N=lane-16, M=r+8
            #pragma unroll
            for (int r = 0; r < 8; ++r) {
                int m = r + hi * 8;
                U[(m * NJ + j) * USTR + m16] = acc[r];
            }
        }
        __syncthreads();

        // ---------------- Phase B: routing for capsule i ----------------
        float l0 = b0[(size_t)i * NJ + j0];
        float l1 = b0[(size_t)i * NJ + j1];
        #pragma unroll
        for (int e = 0; e < NE; ++e) {
            l0 += U[(bb * NJ + j0) * USTR + e] * Vl[(bb * NJ + j0) * USTR + e];
            l1 += U[(bb * NJ + j1) * USTR + e] * Vl[(bb * NJ + j1) * USTR + e];
        }
        // softmax over 32 j's: 16 lanes x 2 each; xor masks 1..8 stay in 16-lane half
        float mx = fmaxf(l0, l1);
        #pragma unroll
        for (int msk = 1; msk <= 8; msk <<= 1)
            mx = fmaxf(mx, __shfl_xor(mx, msk, 32));
        float e0 = __expf(l0 - mx);
        float e1 = __expf(l1 - mx);
        float sm = e0 + e1;
        #pragma unroll
        for (int msk = 1; msk <= 8; msk <<= 1)
            sm += __shfl_xor(sm, msk, 32);
        float inv = 1.0f / sm;
        float c0 = e0 * inv, c1 = e1 * inv;
        #pragma unroll
        for (int e = 0; e < NE; ++e) {
            s0[e] += c0 * U[(bb * NJ + j0) * USTR + e];
            s1[e] += c1 * U[(bb * NJ + j1) * USTR + e];
        }
        __syncthreads();
    }

    // ---- flush partial S ----
    float* sp0 = Sg + ((size_t)(bbase + bb) * NJ + j0) * NE;
    float* sp1 = Sg + ((size_t)(bbase + bb) * NJ + j1) * NE;
    #pragma unroll
    for (int e = 0; e < NE; ++e) {
        atomicAdd(sp0 + e, s0[e]);
        atomicAdd(sp1 + e, s1[e]);
    }
}

// One block per batch, 512 threads = one (j,e) each.
template <int FINAL>
__global__ __launch_bounds__(512) void caps_squash(
    float* __restrict__ S, float* __restrict__ Vacc, float* __restrict__ out)
{
    const int b   = blockIdx.x;
    const int tid = threadIdx.x;           // j = tid>>4, e = tid&15
    const int idx = b * JE + tid;
    float s = S[idx];
    float p = s * s;
    #pragma unroll
    for (int msk = 1; msk <= 8; msk <<= 1)   // reduce over e within 16-lane half
        p += __shfl_xor(p, msk, 32);
    float scale = p / (1.0f + p) * rsqrtf(p + 1e-7f);
    float v = scale * s;
    if (FINAL) {
        out[idx] = v;
    } else {
        Vacc[idx] += v;
        S[idx] = 0.0f;      // reset accumulator for next routing pass
    }
}

extern "C" void kernel_launch(void* const* d_in, const int* in_sizes, int n_in,
                              void* d_out, int out_size, void* d_ws, size_t ws_size,
                              hipStream_t stream) {
    (void)in_sizes; (void)n_in; (void)out_size; (void)ws_size;
    const float* x  = (const float*)d_in[0];
    const float* W  = (const float*)d_in[1];
    const float* b0 = (const float*)d_in[2];
    float* out  = (float*)d_out;
    float* S    = (float*)d_ws;
    float* Vacc = S + BATCH * JE;

    // zero S and Vacc
    caps_zero<<<(2 * BATCH * JE + 255) / 256, 256, 0, stream>>>(S, 2 * BATCH * JE);

    dim3 grid(NI / IT, BATCH / BT);   // 144 x 4 blocks
    // routing iteration 0
    caps_partial<<<grid, 256, 0, stream>>>(x, W, b0, Vacc, S);
    caps_squash<0><<<BATCH, 512, 0, stream>>>(S, Vacc, out);
    // routing iteration 1
    caps_partial<<<grid, 256, 0, stream>>>(x, W, b0, Vacc, S);
    caps_squash<0><<<BATCH, 512, 0, stream>>>(S, Vacc, out);
    // routing iteration 2 (final -> write V)
    caps_partial<<<grid, 256, 0, stream>>>(x, W, b0, Vacc, S);
    caps_squash<1><<<BATCH, 512, 0, stream>>>(S, Vacc, out);
}